// conv_approximation_16939351015451
// MI455X (gfx1250) — compile-verified
//
#include <hip/hip_runtime.h>
#include <hip/hip_bf16.h>
#include <math.h>

// ---------------------------------------------------------------------------
// Performer (FAVOR+) linear attention forward for MI455X (gfx1250, wave32).
// f32 end-to-end; all five GEMM stages (dash2-max, dash2, v2x-update, dash1, out1)
// now run on V_WMMA_F32_16X16X4_F32.
// ---------------------------------------------------------------------------

typedef float v2f __attribute__((ext_vector_type(2)));
typedef float v8f __attribute__((ext_vector_type(8)));

#define B_  16
#define N_  4096
#define H_  8
#define D_  64
#define M_  128
#define BH_ (B_ * H_)

// ws layout in 32-bit words: v2x [BH][M][D], v2sum [BH][M], mxkey [BH] (u32)
#define WS_V2X    0
#define WS_V2SUM  (BH_ * M_ * D_)
#define WS_MXKEY  (WS_V2SUM + BH_ * M_)
#define WS_WORDS  (WS_MXKEY + BH_)

__device__ __forceinline__ unsigned fkey(float f) {
  unsigned u = __float_as_uint(f);
  return (u & 0x80000000u) ? ~u : (u | 0x80000000u);   // order-preserving map
}
__device__ __forceinline__ float fdec(unsigned k) {
  return (k & 0x80000000u) ? __uint_as_float(k & 0x7fffffffu)
                           : __uint_as_float(~k);
}

constexpr float S_SCALE = 0.35355339059327373f;  // tau^-1/2 * d^-1/4 (d=64)
constexpr float RATIO   = 0.08838834764831845f;  // 1/sqrt(m), m=128
constexpr float EPS_    = 1e-6f;

__device__ __forceinline__ v8f wmma4(v2f a, v2f b, v8f c) {
  // D = A(16x4,f32) * B(4x16,f32) + C(16x16,f32)
  return __builtin_amdgcn_wmma_f32_16x16x4_f32(false, a, false, b, (short)0,
                                               c, false, false);
}

// ---------------------------------------------------------------- k0: init ws
__global__ void perf_k0_init(unsigned* __restrict__ ws) {
  const size_t i = (size_t)blockIdx.x * 256 + threadIdx.x;
  if (i < (size_t)WS_WORDS) ws[i] = 0u;   // key 0 decodes to -max for mx
}

// -------------------------------------- k1: global max of dash2 (WMMA GEMM)
// grid: BH*32 blocks (128 rows each), 128 threads = 4 waves
__global__ __launch_bounds__(128) void perf_k1_max2(
    const float* __restrict__ nv2, const float* __restrict__ proj,
    unsigned* __restrict__ mxkey) {
  __shared__ float projs[M_ * D_];
  __shared__ float atile[16 * D_];
  __shared__ float red[128];
  const int bh = blockIdx.x >> 5, chunk = blockIdx.x & 31;
  const int b = bh >> 3, h = bh & 7;
  const int t = threadIdx.x, lane = t & 31, wave = t >> 5;

  for (int i = t; i < M_ * D_ / 4; i += 128)
    ((float4*)projs)[i] = ((const float4*)proj)[i];

  const int r = t >> 3, c8 = (t & 7) * 8;
  const int Mr = lane & 15, kh = (lane >> 4) * 2;
  const int ct0 = wave * 2, ct1 = ct0 + 1;

  float bmx = -3.4e38f;
  for (int tile = 0; tile < 8; ++tile) {
    const int n0 = chunk * 128 + tile * 16;
    {
      const size_t off = ((size_t)((b * N_ + n0 + r) * H_ + h)) * (size_t)D_ + c8;
      float4 a0 = *(const float4*)(nv2 + off);
      float4 a1 = *(const float4*)(nv2 + off + 4);
      atile[r * D_ + c8 + 0] = a0.x * S_SCALE; atile[r * D_ + c8 + 1] = a0.y * S_SCALE;
      atile[r * D_ + c8 + 2] = a0.z * S_SCALE; atile[r * D_ + c8 + 3] = a0.w * S_SCALE;
      atile[r * D_ + c8 + 4] = a1.x * S_SCALE; atile[r * D_ + c8 + 5] = a1.y * S_SCALE;
      atile[r * D_ + c8 + 6] = a1.z * S_SCALE; atile[r * D_ + c8 + 7] = a1.w * S_SCALE;
    }
    __syncthreads();

    v8f c0 = {0, 0, 0, 0, 0, 0, 0, 0}, c1 = {0, 0, 0, 0, 0, 0, 0, 0};
#pragma unroll
    for (int kk = 0; kk < 16; ++kk) {
      const int koff = kk * 4 + kh;
      v2f a;  a.x = atile[Mr * D_ + koff];               a.y = atile[Mr * D_ + koff + 1];
      v2f b0; b0.x = projs[(ct0 * 16 + Mr) * D_ + koff]; b0.y = projs[(ct0 * 16 + Mr) * D_ + koff + 1];
      v2f b1; b1.x = projs[(ct1 * 16 + Mr) * D_ + koff]; b1.y = projs[(ct1 * 16 + Mr) * D_ + koff + 1];
      c0 = wmma4(a, b0, c0);
      c1 = wmma4(a, b1, c1);
    }
#pragma unroll
    for (int j = 0; j < 8; ++j) bmx = fmaxf(bmx, fmaxf(c0[j], c1[j]));
    __syncthreads();
  }

  red[t] = bmx;
  __syncthreads();
  for (int s = 64; s > 0; s >>= 1) {
    if (t < s) red[t] = fmaxf(red[t], red[t + s]);
    __syncthreads();
  }
  if (t == 0) atomicMax(mxkey + bh, fkey(red[0]));
}

// ------ k2: v2p (WMMA dash2 + exp) + v2x rank-16 WMMA updates + v2sum
// grid: BH*16 blocks (256 rows each), 128 threads = 4 waves
__global__ __launch_bounds__(128) void perf_k2_v2(
    const float* __restrict__ nv2, const float* __restrict__ x,
    const float* __restrict__ proj, const unsigned* __restrict__ mxkey,
    float* __restrict__ v2p_out, float* __restrict__ v2x_ws,
    float* __restrict__ v2sum_ws) {
  __shared__ float projs[M_ * D_];
  __shared__ float atile[16 * D_];
  __shared__ float xtile[16 * D_];
  __shared__ float vtile[16 * M_];
  __shared__ float diag[16];

  const int bh = blockIdx.x >> 4, chunk = blockIdx.x & 15;
  const int b = bh >> 3, h = bh & 7;
  const int t = threadIdx.x, lane = t & 31, wave = t >> 5;

  for (int i = t; i < M_ * D_ / 4; i += 128)
    ((float4*)projs)[i] = ((const float4*)proj)[i];
  const float mx2 = fdec(mxkey[bh]);

  const int r = t >> 3, c8 = (t & 7) * 8;
  const int Mr = lane & 15;                // row (A) / col (B, C/D) index
  const int kh = (lane >> 4) * 2;          // K sub-offset for A/B frags
  const int Mbase = (lane >> 4) * 8;       // C/D row base
  const int ct0 = wave * 2, ct1 = ct0 + 1; // m-column tiles for this wave

  // persistent v2x partials: wave covers m-tiles {ct0, ct1} x 4 d-tiles
  v8f acc0[4], acc1[4];
#pragma unroll
  for (int dt = 0; dt < 4; ++dt) {
    acc0[dt] = (v8f){0, 0, 0, 0, 0, 0, 0, 0};
    acc1[dt] = (v8f){0, 0, 0, 0, 0, 0, 0, 0};
  }
  float vsr = 0.f;

  for (int tile = 0; tile < 16; ++tile) {
    const int n0 = chunk * 256 + tile * 16;
    const size_t off = ((size_t)((b * N_ + n0 + r) * H_ + h)) * (size_t)D_ + c8;
    {
      float4 a0 = *(const float4*)(nv2 + off);
      float4 a1 = *(const float4*)(nv2 + off + 4);
      atile[r * D_ + c8 + 0] = a0.x * S_SCALE; atile[r * D_ + c8 + 1] = a0.y * S_SCALE;
      atile[r * D_ + c8 + 2] = a0.z * S_SCALE; atile[r * D_ + c8 + 3] = a0.w * S_SCALE;
      atile[r * D_ + c8 + 4] = a1.x * S_SCALE; atile[r * D_ + c8 + 5] = a1.y * S_SCALE;
      atile[r * D_ + c8 + 6] = a1.z * S_SCALE; atile[r * D_ + c8 + 7] = a1.w * S_SCALE;
      float4 x0 = *(const float4*)(x + off);
      float4 x1 = *(const float4*)(x + off + 4);
      xtile[r * D_ + c8 + 0] = x0.x; xtile[r * D_ + c8 + 1] = x0.y;
      xtile[r * D_ + c8 + 2] = x0.z; xtile[r * D_ + c8 + 3] = x0.w;
      xtile[r * D_ + c8 + 4] = x1.x; xtile[r * D_ + c8 + 5] = x1.y;
      xtile[r * D_ + c8 + 6] = x1.z; xtile[r * D_ + c8 + 7] = x1.w;
    }
    if (tile < 15) {  // speculative prefetch of next tile's rows
      __builtin_prefetch(nv2 + off + (size_t)16 * H_ * D_, 0, 1);
      __builtin_prefetch(x + off + (size_t)16 * H_ * D_, 0, 1);
    }
    __syncthreads();
    if (t < 16) {
      float s = 0.f;
#pragma unroll
      for (int k = 0; k < D_; ++k) { float v = atile[t * D_ + k]; s += v * v; }
      diag[t] = 0.5f * s;
    }
    __syncthreads();

    // dash2 = scaled_nv2 . proj^T  (K = 64)
    v8f c0 = {0, 0, 0, 0, 0, 0, 0, 0}, c1 = {0, 0, 0, 0, 0, 0, 0, 0};
#pragma unroll
    for (int kk = 0; kk < 16; ++kk) {
      const int koff = kk * 4 + kh;
      v2f a;  a.x = atile[Mr * D_ + koff];               a.y = atile[Mr * D_ + koff + 1];
      v2f b0; b0.x = projs[(ct0 * 16 + Mr) * D_ + koff]; b0.y = projs[(ct0 * 16 + Mr) * D_ + koff + 1];
      v2f b1; b1.x = projs[(ct1 * 16 + Mr) * D_ + koff]; b1.y = projs[(ct1 * 16 + Mr) * D_ + koff + 1];
      c0 = wmma4(a, b0, c0);
      c1 = wmma4(a, b1, c1);
    }
#pragma unroll
    for (int j = 0; j < 8; ++j) {
      const int Mrr = j + Mbase;
      vtile[Mrr * M_ + ct0 * 16 + Mr] = RATIO * (__expf(c0[j] - diag[Mrr] - mx2) + EPS_);
      vtile[Mrr * M_ + ct1 * 16 + Mr] = RATIO * (__expf(c1[j] - diag[Mrr] - mx2) + EPS_);
    }
    __syncthreads();

    {  // coalesced v2p store + v2sum partial (thread t owns m-column t)
      const size_t obase = ((size_t)((b * N_ + n0) * H_ + h)) * (size_t)M_ + t;
      for (int rr = 0; rr < 16; ++rr) {
        const float p = vtile[rr * M_ + t];
        v2p_out[obase + (size_t)rr * (H_ * M_)] = p;
        vsr += p;
      }
    }

    // v2x rank-16 update: A = v2p^T (m x rows), B = x tile (rows x d), K = 16
#pragma unroll
    for (int kk = 0; kk < 4; ++kk) {
      const int koff = kk * 4 + kh;   // K index = row within tile
      v2f a0; a0.x = vtile[koff * M_ + ct0 * 16 + Mr]; a0.y = vtile[(koff + 1) * M_ + ct0 * 16 + Mr];
      v2f a1; a1.x = vtile[koff * M_ + ct1 * 16 + Mr]; a1.y = vtile[(koff + 1) * M_ + ct1 * 16 + Mr];
#pragma unroll
      for (int dt = 0; dt < 4; ++dt) {
        v2f bx; bx.x = xtile[koff * D_ + dt * 16 + Mr]; bx.y = xtile[(koff + 1) * D_ + dt * 16 + Mr];
        acc0[dt] = wmma4(a0, bx, acc0[dt]);
        acc1[dt] = wmma4(a1, bx, acc1[dt]);
      }
    }
    __syncthreads();
  }

  // flush partial v2x (C frag element: m = mt*16 + j + Mbase, d = dt*16 + Mr)
  float* vxb = v2x_ws + (size_t)bh * (M_ * D_);
#pragma unroll
  for (int dt = 0; dt < 4; ++dt) {
#pragma unroll
    for (int j = 0; j < 8; ++j) {
      atomicAdd(vxb + (ct0 * 16 + j + Mbase) * D_ + dt * 16 + Mr, acc0[dt][j]);
      atomicAdd(vxb + (ct1 * 16 + j + Mbase) * D_ + dt * 16 + Mr, acc1[dt][j]);
    }
  }
  atomicAdd(v2sum_ws + bh * M_ + t, vsr);
}

// --------------------------------- k3a: v1p (WMMA dash1 + rowmax + exp)
// grid: BH*32 blocks (128 rows each), 128 threads = 4 waves
__global__ __launch_bounds__(128) void perf_k3a_v1(
    const float* __restrict__ nv1, const float* __restrict__ proj,
    float* __restrict__ v1p_out) {
  __shared__ float projs[M_ * D_];
  __shared__ float atile[16 * D_];
  __shared__ float dtile[16 * M_];
  __shared__ float diag[16];
  __shared__ float rmax[16];

  const int bh = blockIdx.x >> 5, chunk = blockIdx.x & 31;
  const int b = bh >> 3, h = bh & 7;
  const int t = threadIdx.x, lane = t & 31, wave = t >> 5;

  for (int i = t; i < M_ * D_ / 4; i += 128)
    ((float4*)projs)[i] = ((const float4*)proj)[i];

  const int r = t >> 3, c8 = (t & 7) * 8;
  const int Mr = lane & 15, kh = (lane >> 4) * 2, Mbase = (lane >> 4) * 8;
  const int ct0 = wave * 2, ct1 = ct0 + 1;

  for (int tile = 0; tile < 8; ++tile) {
    const int n0 = chunk * 128 + tile * 16;
    {
      const size_t off = ((size_t)((b * N_ + n0 + r) * H_ + h)) * (size_t)D_ + c8;
      float4 a0 = *(const float4*)(nv1 + off);
      float4 a1 = *(const float4*)(nv1 + off + 4);
      atile[r * D_ + c8 + 0] = a0.x * S_SCALE; atile[r * D_ + c8 + 1] = a0.y * S_SCALE;
      atile[r * D_ + c8 + 2] = a0.z * S_SCALE; atile[r * D_ + c8 + 3] = a0.w * S_SCALE;
      atile[r * D_ + c8 + 4] = a1.x * S_SCALE; atile[r * D_ + c8 + 5] = a1.y * S_SCALE;
      atile[r * D_ + c8 + 6] = a1.z * S_SCALE; atile[r * D_ + c8 + 7] = a1.w * S_SCALE;
    }
    __syncthreads();
    if (t < 16) {
      float s = 0.f;
#pragma unroll
      for (int k = 0; k < D_; ++k) { float v = atile[t * D_ + k]; s += v * v; }
      diag[t] = 0.5f * s;
    }
    __syncthreads();

    v8f c0 = {0, 0, 0, 0, 0, 0, 0, 0}, c1 = {0, 0, 0, 0, 0, 0, 0, 0};
#pragma unroll
    for (int kk = 0; kk < 16; ++kk) {
      const int koff = kk * 4 + kh;
      v2f a;  a.x = atile[Mr * D_ + koff];               a.y = atile[Mr * D_ + koff + 1];
      v2f b0; b0.x = projs[(ct0 * 16 + Mr) * D_ + koff]; b0.y = projs[(ct0 * 16 + Mr) * D_ + koff + 1];
      v2f b1; b1.x = projs[(ct1 * 16 + Mr) * D_ + koff]; b1.y = projs[(ct1 * 16 + Mr) * D_ + koff + 1];
      c0 = wmma4(a, b0, c0);
      c1 = wmma4(a, b1, c1);
    }
#pragma unroll
    for (int j = 0; j < 8; ++j) {
      const int Mrr = j + Mbase;
      dtile[Mrr * M_ + ct0 * 16 + Mr] = c0[j];
      dtile[Mrr * M_ + ct1 * 16 + Mr] = c1[j];
    }
    __syncthreads();
    if (t < 16) {
      float m = -3.4e38f;
      for (int c = 0; c < M_; ++c) m = fmaxf(m, dtile[t * M_ + c]);
      rmax[t] = m;
    }
    __syncthreads();
    {
      const size_t obase = ((size_t)((b * N_ + n0) * H_ + h)) * (size_t)M_ + t;
      for (int rr = 0; rr < 16; ++rr) {
        const float val = RATIO * (__expf(dtile[rr * M_ + t] - diag[rr] - rmax[rr]) + EPS_);
        v1p_out[obase + (size_t)rr * (H_ * M_)] = val;
      }
    }
    __syncthreads();
  }
}

// ----------------------- k3b: out = (v1p @ v2x) / (v1p . v2sum)  (WMMA K=128)
// grid: BH*32 blocks (128 rows each), 128 threads = 4 waves (wave -> D tile)
__global__ __launch_bounds__(128) void perf_k3b_out(
    const float* __restrict__ v1p, const float* __restrict__ v2x_ws,
    const float* __restrict__ v2sum_ws, float* __restrict__ out) {
  __shared__ float v2xl[M_ * D_];
  __shared__ float v2s[M_];
  __shared__ float ptile[16 * M_];
  __shared__ float inv2[16];

  const int bh = blockIdx.x >> 5, chunk = blockIdx.x & 31;
  const int b = bh >> 3, h = bh & 7;
  const int t = threadIdx.x, lane = t & 31, wave = t >> 5;

  for (int i = t; i < M_ * D_ / 4; i += 128)
    ((float4*)v2xl)[i] = ((const float4*)(v2x_ws + (size_t)bh * (M_ * D_)))[i];
  if (t < M_) v2s[t] = v2sum_ws[bh * M_ + t];

  const int Mr = lane & 15, kh = (lane >> 4) * 2, Mbase = (lane >> 4) * 8;
  const int ct = wave;  // D column tile (D=64 -> 4 tiles of 16)

  for (int tile = 0; tile < 8; ++tile) {
    const int n0 = chunk * 128 + tile * 16;
    for (int i = t; i < 16 * M_ / 4; i += 128) {
      const int rr = i >> 5, c4 = i & 31;
      ((float4*)ptile)[i] =
          *(const float4*)(v1p + ((size_t)((b * N_ + n0 + rr) * H_ + h)) * (size_t)M_ + c4 * 4);
    }
    __syncthreads();
    if (t < 16) {
      float s = 0.f;
#pragma unroll
      for (int k = 0; k < M_; ++k) s += ptile[t * M_ + k] * v2s[k];
      inv2[t] = 1.0f / s;
    }
    __syncthreads();

    v8f cc = {0, 0, 0, 0, 0, 0, 0, 0};
#pragma unroll
    for (int kk = 0; kk < 32; ++kk) {
      const int koff = kk * 4 + kh;
      v2f a;  a.x = ptile[Mr * M_ + koff];           a.y = ptile[Mr * M_ + koff + 1];
      v2f b2; b2.x = v2xl[koff * D_ + ct * 16 + Mr]; b2.y = v2xl[(koff + 1) * D_ + ct * 16 + Mr];
      cc = wmma4(a, b2, cc);
    }
#pragma unroll
    for (int j = 0; j < 8; ++j) {
      const int Mrr = j + Mbase;
      out[((size_t)((b * N_ + n0 + Mrr) * H_ + h)) * (size_t)D_ + ct * 16 + Mr] =
          cc[j] * inv2[Mrr];
    }
    __syncthreads();
  }
}

// ---------------------------------------------------------------------------
extern "C" void kernel_launch(void* const* d_in, const int* in_sizes, int n_in,
                              void* d_out, int out_size, void* d_ws, size_t ws_size,
                              hipStream_t stream) {
  (void)in_sizes; (void)n_in; (void)out_size; (void)ws_size;
  const float* x    = (const float*)d_in[0];
  const float* nv1  = (const float*)d_in[1];
  const float* nv2  = (const float*)d_in[2];
  const float* proj = (const float*)d_in[3];

  float* out1 = (float*)d_out;                              // [B,N,H,D]
  float* v1p  = out1 + (size_t)B_ * N_ * H_ * D_;           // [B,N,H,M]
  float* v2p  = v1p + (size_t)B_ * N_ * H_ * M_;            // [B,N,H,M]

  float* ws       = (float*)d_ws;
  float* v2x_ws   = ws + WS_V2X;
  float* v2sum_ws = ws + WS_V2SUM;
  unsigned* mxkey = (unsigned*)(ws + WS_MXKEY);

  perf_k0_init<<<(WS_WORDS + 255) / 256, 256, 0, stream>>>((unsigned*)d_ws);
  perf_k1_max2<<<BH_ * 32, 128, 0, stream>>>(nv2, proj, mxkey);
  perf_k2_v2<<<BH_ * 16, 128, 0, stream>>>(nv2, x, proj, mxkey, v2p, v2x_ws, v2sum_ws);
  perf_k3a_v1<<<BH_ * 32, 128, 0, stream>>>(nv1, proj, v1p);
  perf_k3b_out<<<BH_ * 32, 128, 0, stream>>>(v1p, v2x_ws, v2sum_ws, out1);
}